// VQVAE_52295521796299
// MI455X (gfx1250) — compile-verified
//
#include <hip/hip_runtime.h>
#include <hip/hip_bf16.h>

// VQ-VAE codebook step for MI455X (gfx1250, wave32).
// B=256, D=256, K=1024, E=16. fp32 throughout (matches reference argmin exactly).
// Bound by the 268MB one_hot write -> nontemporal stores; matmul via v_wmma_f32_16x16x4_f32.

#define Bb 256
#define Dd 256
#define Kk 1024
#define Ee 16
#define DECAYf 0.999f
#define GAINf  (1.0f - 0.999f)
#define EPSf   1e-06f

typedef __attribute__((ext_vector_type(2))) float v2f;
typedef __attribute__((ext_vector_type(4))) float v4f;
typedef __attribute__((ext_vector_type(8))) float v8f;

// ---------------------------------------------------------------------------
// Kernel A: per (d, batch-tile-of-16) distance + argmin + cw_embed + one_hot.
// 256 threads = 8 waves; wave w owns K-slice [w*128, w*128+128).
// dist ranking uses c2[k] - 2*xc (x2 term is row-constant, drops out of argmin).
// ---------------------------------------------------------------------------
__global__ __launch_bounds__(256) void vq_argmin_kernel(
    const float* __restrict__ cw_q,      // (B, D*E)
    const float* __restrict__ codebook,  // (D, K, E)
    float* __restrict__ out_embed,       // (B, D*E)
    float* __restrict__ out_onehot,      // (B, D, K)
    int*   __restrict__ ws_idx)          // (B, D)
{
    const int d    = blockIdx.x;
    const int b0   = blockIdx.y * 16;
    const int t    = threadIdx.x;
    const int wave = t >> 5;
    const int lane = t & 31;
    const int hv   = lane >> 4;   // half-wave select
    const int lm   = lane & 15;

    __shared__ float xs[16 * 17];     // x tile, padded rows (17 floats) vs bank conflicts
    __shared__ float c2s[Kk];         // |c|^2 per code
    __shared__ float redV[16][8];     // per-row per-wave best value
    __shared__ int   redI[16][8];     // per-row per-wave best index
    __shared__ int   fidx[16];        // final argmin per batch row

    const float* __restrict__ cbd = codebook + (size_t)d * (Kk * Ee);

    // Stage x tile: rows b0..b0+15, dims d*16..d*16+15
    {
        int m = t >> 4, e = t & 15;
        xs[m * 17 + e] = cw_q[(size_t)(b0 + m) * (Dd * Ee) + d * Ee + e];
    }
    // c2[k] = sum_e codebook[d][k][e]^2   (4 codes per thread)
    for (int s = 0; s < 4; ++s) {
        int k = t + s * 256;
        const float* row = cbd + (size_t)k * Ee;
        float acc = 0.f;
#pragma unroll
        for (int e = 0; e < 16; ++e) { float v = row[e]; acc += v * v; }
        c2s[k] = acc;
    }
    __syncthreads();

    // A fragments: 16x4 f32 per WMMA. Lane layout (ISA 7.12.2):
    //   lanes 0-15: M=lane, K = {e0, e0+1};  lanes 16-31: M=lane-16, K = {e0+2, e0+3}
    const int eoff = 2 * hv;
    v2f a0, a1, a2, a3;
    a0[0] = xs[lm * 17 + 0 + eoff];  a0[1] = xs[lm * 17 + 1 + eoff];
    a1[0] = xs[lm * 17 + 4 + eoff];  a1[1] = xs[lm * 17 + 5 + eoff];
    a2[0] = xs[lm * 17 + 8 + eoff];  a2[1] = xs[lm * 17 + 9 + eoff];
    a3[0] = xs[lm * 17 + 12 + eoff]; a3[1] = xs[lm * 17 + 13 + eoff];

    float best[8];
    int   bidx[8];
#pragma unroll
    for (int i = 0; i < 8; ++i) { best[i] = 3.4e38f; bidx[i] = 0; }

    // Sweep this wave's 128 codes, 16 at a time
    for (int t8 = 0; t8 < 8; ++t8) {
        const int kbase = wave * 128 + t8 * 16;
        const int kk    = kbase + lm;
        // B fragments (4x16 f32): lane's column N=lm, rows K={e0+2*hv, e0+2*hv+1}
        const float* brow = cbd + (size_t)kk * Ee + eoff;   // 8B aligned
        v2f bv0 = *(const v2f*)(brow + 0);
        v2f bv1 = *(const v2f*)(brow + 4);
        v2f bv2 = *(const v2f*)(brow + 8);
        v2f bv3 = *(const v2f*)(brow + 12);

        v8f acc = {};
        acc = __builtin_amdgcn_wmma_f32_16x16x4_f32(false, a0, false, bv0, (short)0, acc, false, false);
        acc = __builtin_amdgcn_wmma_f32_16x16x4_f32(false, a1, false, bv1, (short)0, acc, false, false);
        acc = __builtin_amdgcn_wmma_f32_16x16x4_f32(false, a2, false, bv2, (short)0, acc, false, false);
        acc = __builtin_amdgcn_wmma_f32_16x16x4_f32(false, a3, false, bv3, (short)0, acc, false, false);

        const float c2v = c2s[kk];
#pragma unroll
        for (int i = 0; i < 8; ++i) {
            float dv = c2v - 2.0f * acc[i];
            if (dv < best[i] || (dv == best[i] && kk < bidx[i])) { best[i] = dv; bidx[i] = kk; }
        }
    }

    // Lane reduction within each 16-lane half (columns), rows m = hv*8 + i
#pragma unroll
    for (int i = 0; i < 8; ++i) {
        float bvv = best[i]; int bii = bidx[i];
        for (int off = 8; off >= 1; off >>= 1) {
            float ov = __shfl_xor(bvv, off, 32);
            int   oi = __shfl_xor(bii, off, 32);
            if (ov < bvv || (ov == bvv && oi < bii)) { bvv = ov; bii = oi; }
        }
        if (lm == 0) { int m = hv * 8 + i; redV[m][wave] = bvv; redI[m][wave] = bii; }
    }
    __syncthreads();

    // Cross-wave reduction: thread t < 16 owns batch row t
    if (t < 16) {
        float bvv = redV[t][0]; int bii = redI[t][0];
        for (int w = 1; w < 8; ++w) {
            float ov = redV[t][w]; int oi = redI[t][w];
            if (ov < bvv || (ov == bvv && oi < bii)) { bvv = ov; bii = oi; }
        }
        fidx[t] = bii;
        ws_idx[(size_t)(b0 + t) * Dd + d] = bii;
    }
    __syncthreads();

    // cw_embed gather: out[b][d*E+e] = codebook[d][idx][e]
    {
        int m = t >> 4, e = t & 15;
        out_embed[(size_t)(b0 + m) * (Dd * Ee) + d * Ee + e] =
            cbd[(size_t)fidx[m] * Ee + e];
    }

    // one_hot: 16 rows x 1024 cols, 16 threads per row, 64 cols each.
    // 268MB total across grid -> nontemporal float4 stores (don't thrash L2).
    {
        int m    = t >> 4;
        int col0 = (t & 15) * 64;
        int id   = fidx[m];
        float* base = out_onehot + ((size_t)(b0 + m) * Dd + d) * Kk;
        for (int c = col0; c < col0 + 64; c += 4) {
            v4f v;
            v[0] = (id == c + 0) ? 1.0f : 0.0f;
            v[1] = (id == c + 1) ? 1.0f : 0.0f;
            v[2] = (id == c + 2) ? 1.0f : 0.0f;
            v[3] = (id == c + 3) ? 1.0f : 0.0f;
            __builtin_nontemporal_store(v, (v4f*)(base + c));
        }
    }
}

// ---------------------------------------------------------------------------
// Kernel B: EMA updates. One block per d; gather-reduce over batch (no atomics,
// deterministic). Each thread owns 4 codes; idx + x staged in LDS.
// ---------------------------------------------------------------------------
__global__ __launch_bounds__(256) void vq_update_kernel(
    const float* __restrict__ cw_q,      // (B, D*E)
    const float* __restrict__ codebook,  // (D, K, E)
    const float* __restrict__ ema,       // (D, K)
    const int*   __restrict__ ws_idx,    // (B, D)
    float* __restrict__ new_cb,          // (D, K, E)
    float* __restrict__ new_ema)         // (D, K)
{
    const int d = blockIdx.x;
    const int t = threadIdx.x;

    __shared__ int   idxs[Bb];
    __shared__ float xloc[Bb][17];

    idxs[t] = ws_idx[(size_t)t * Dd + d];
    {
        const float* xr = cw_q + (size_t)t * (Dd * Ee) + d * Ee;
#pragma unroll
        for (int e = 0; e < 16; ++e) xloc[t][e] = xr[e];
    }
    __syncthreads();

    for (int s = 0; s < 4; ++s) {
        const int k = s * 256 + t;
        float acc[16];
#pragma unroll
        for (int e = 0; e < 16; ++e) acc[e] = 0.f;
        float cnt = 0.f;
        for (int b = 0; b < Bb; ++b) {
            if (idxs[b] == k) {       // idxs[b] uniform across the wave -> broadcast
                cnt += 1.0f;
#pragma unroll
                for (int e = 0; e < 16; ++e) acc[e] += xloc[b][e];
            }
        }
        const size_t dk = (size_t)d * Kk + k;
        const float ev  = ema[dk];
        const float inv = GAINf / (ev + EPSf);
        const float* crow = codebook + dk * Ee;
        float*       orow = new_cb   + dk * Ee;
#pragma unroll
        for (int e = 0; e < 16; ++e) orow[e] = crow[e] * DECAYf + acc[e] * inv;
        new_ema[dk] = DECAYf * ev + GAINf * cnt;
    }
}

extern "C" void kernel_launch(void* const* d_in, const int* in_sizes, int n_in,
                              void* d_out, int out_size, void* d_ws, size_t ws_size,
                              hipStream_t stream) {
    const float* cw_q     = (const float*)d_in[0];   // (B, D*E)
    const float* codebook = (const float*)d_in[1];   // (D, K, E)
    const float* ema      = (const float*)d_in[2];   // (D, K)

    float* out       = (float*)d_out;
    float* out_embed = out;                                    // B*D*E
    float* out_oh    = out_embed + (size_t)Bb * Dd * Ee;       // B*D*K
    float* out_cb    = out_oh    + (size_t)Bb * Dd * Kk;       // D*K*E
    float* out_ema   = out_cb    + (size_t)Dd * Kk * Ee;       // D*K

    int* ws_idx = (int*)d_ws;   // needs B*D*4 = 256 KB scratch

    dim3 gridA(Dd, Bb / 16);    // (256, 16)
    vq_argmin_kernel<<<gridA, 256, 0, stream>>>(cw_q, codebook, out_embed, out_oh, ws_idx);
    vq_update_kernel<<<Dd, 256, 0, stream>>>(cw_q, codebook, ema, ws_idx, out_cb, out_ema);
    (void)in_sizes; (void)n_in; (void)out_size; (void)ws_size;
}